// GAT_15479062135293
// MI455X (gfx1250) — compile-verified
//
#include <hip/hip_runtime.h>
#include <hip/hip_bf16.h>
#include <math.h>

// ---------------------------------------------------------------------------
// GAT (3-layer) for MI455X / gfx1250, wave32.
//  - Dense projections x@W run on the WMMA pipe: V_WMMA_F32_16X16X4_F32
//    (fp32 in/out, 16x16 tile per wave, K-loop over 128). Out-of-range lanes
//    are index-clamped (not predicated) so the K-loop has no EXEC churn.
//  - Edge softmax + aggregation are bandwidth-bound streaming passes:
//    wave-per-edge float4 gathers and f32 atomic scatters (L2-resident dst).
// ---------------------------------------------------------------------------

typedef float v2f __attribute__((ext_vector_type(2)));
typedef float v8f __attribute__((ext_vector_type(8)));

#define NEG_SLOPE 0.2f

// float atomic max via ordered-int mapping (init must be -inf)
__device__ __forceinline__ void atomicMaxF(float* addr, float v) {
  if (v >= 0.0f) {
    atomicMax((int*)addr, __float_as_int(v));
  } else {
    atomicMin((unsigned int*)addr, __float_as_uint(v));
  }
}

// --------------------------- WMMA fp32 GEMM --------------------------------
// C[N x Fout] = A[N x K] * B[K x Fout], one 16x16 output tile per wave32.
// A 16x4 fp32 layout : lanes 0-15 -> M=lane, K pair 0-1 ; lanes 16-31 -> K pair 2-3
// B 4x16  fp32 layout : lanes 0-15 -> N=lane, K pair 0-1 ; lanes 16-31 -> K pair 2-3
// C/D 16x16 fp32      : VGPR i, lanes 0-15 -> M=i ; lanes 16-31 -> M=i+8
// Out-of-range rows/cols: clamp index (result column/row is a duplicate that
// is simply not stored) -> branch-free inner loop, EXEC stays all-ones.
__global__ void k_gemm_wmma(const float* __restrict__ A, const float* __restrict__ B,
                            float* __restrict__ Cg, int Nrows, int K, int Fout,
                            int tiles_m, int tiles_n) {
  const int wid  = (blockIdx.x * blockDim.x + threadIdx.x) >> 5;
  const int lane = threadIdx.x & 31;
  if (wid >= tiles_m * tiles_n) return;
  const int tm   = wid / tiles_n;
  const int tn   = wid - tm * tiles_n;
  const int half = lane >> 4;      // 0: K pair 0-1, 1: K pair 2-3
  const int l15  = lane & 15;
  const int arow = tm * 16 + l15;  // M index for A fragment
  const int bcol = tn * 16 + l15;  // N index for B fragment
  const bool bvalid = bcol < Fout;
  const int arowc = (arow < Nrows) ? arow : (Nrows - 1);
  const int bcolc = bvalid ? bcol : (Fout - 1);

  const float* aptr = A + (size_t)arowc * K + half * 2;           // +4 per step
  const float* bptr = B + (size_t)(half * 2) * Fout + bcolc;      // +4*Fout per step
  const size_t bstep = (size_t)4 * (size_t)Fout;

  v8f acc = {};
#pragma unroll 4
  for (int k = 0; k < K; k += 4) {
    const v2f a = *(const v2f*)aptr;       // contiguous K pair (8B aligned)
    v2f b;
    b.x = bptr[0];
    b.y = bptr[Fout];
    // 8 args: (neg_a, A, neg_b, B, c_mod, C, reuse_a, reuse_b)
    acc = __builtin_amdgcn_wmma_f32_16x16x4_f32(false, a, false, b,
                                                (short)0, acc, false, false);
    aptr += 4;
    bptr += bstep;
  }

  if (bvalid) {
    const int baserow = tm * 16 + half * 8;
#pragma unroll
    for (int i = 0; i < 8; ++i) {
      const int r = baserow + i;
      if (r < Nrows) Cg[(size_t)r * Fout + bcol] = acc[i];
    }
  }
}

// ------------------------ attention logit reduce ---------------------------
// als[n,h] = sum_c h[n,h,c]*a_src[h,c] ; ald likewise.
__global__ void k_logits(const float* __restrict__ h, const float* __restrict__ a_src,
                         const float* __restrict__ a_dst, float* __restrict__ als,
                         float* __restrict__ ald, int N, int H, int C) {
  const int idx = blockIdx.x * blockDim.x + threadIdx.x;
  if (idx >= N * H) return;
  const int n = idx / H, hh = idx - n * H;
  const float* hp = h + (size_t)n * H * C + (size_t)hh * C;
  const float* ws = a_src + hh * C;
  const float* wd = a_dst + hh * C;
  float ss = 0.0f, sd = 0.0f;
  for (int c = 0; c < C; ++c) {
    const float v = hp[c];
    ss += v * ws[c];
    sd += v * wd[c];
  }
  als[idx] = ss;
  ald[idx] = sd;
}

__global__ void k_fill(float* __restrict__ p, float v, int n) {
  const int idx = blockIdx.x * blockDim.x + threadIdx.x;
  if (idx < n) p[idx] = v;
}

// -------------------------- edge softmax passes ----------------------------
__device__ __forceinline__ void edge_sd(int e, int E, const int* __restrict__ src,
                                        const int* __restrict__ dst, int* s, int* d) {
  if (e < E) { *s = src[e]; *d = dst[e]; }
  else       { *s = e - E;  *d = e - E; }   // appended self-loops
}

__global__ void k_edge_max(const float* __restrict__ als, const float* __restrict__ ald,
                           const int* __restrict__ src, const int* __restrict__ dst,
                           float* __restrict__ emax, int E, int Etot, int H) {
  const int idx = blockIdx.x * blockDim.x + threadIdx.x;
  if (idx >= Etot * H) return;
  const int e = idx / H, hh = idx - e * H;
  int s, d; edge_sd(e, E, src, dst, &s, &d);
  float z = als[s * H + hh] + ald[d * H + hh];
  z = (z > 0.0f) ? z : NEG_SLOPE * z;
  atomicMaxF(&emax[d * H + hh], z);
}

__global__ void k_edge_expsum(const float* __restrict__ als, const float* __restrict__ ald,
                              const int* __restrict__ src, const int* __restrict__ dst,
                              const float* __restrict__ emax, float* __restrict__ den,
                              float* __restrict__ exb, int E, int Etot, int H) {
  const int idx = blockIdx.x * blockDim.x + threadIdx.x;
  if (idx >= Etot * H) return;
  const int e = idx / H, hh = idx - e * H;
  int s, d; edge_sd(e, E, src, dst, &s, &d);
  float z = als[s * H + hh] + ald[d * H + hh];
  z = (z > 0.0f) ? z : NEG_SLOPE * z;
  float m = emax[d * H + hh];
  if (!isfinite(m)) m = 0.0f;                 // mirrors reference where()
  const float ev = __expf(z - m);
  exb[idx] = ev;
  atomicAdd(&den[d * H + hh], ev);
}

// den -> 1/(den + eps), so aggregation does mul instead of div per edge.
__global__ void k_recip(float* __restrict__ den, int n) {
  const int idx = blockIdx.x * blockDim.x + threadIdx.x;
  if (idx < n) den[idx] = 1.0f / (den[idx] + 1e-16f);
}

// ----------------------------- aggregation ---------------------------------
// F=128, H=8, C=16: one wave per edge; lane owns 4 contiguous floats, so the
// 32 lanes cover exactly one 512B feature row (coalesced float4 gather).
__global__ void k_agg128(const float* __restrict__ h, const float* __restrict__ exb,
                         const float* __restrict__ rden, const int* __restrict__ src,
                         const int* __restrict__ dst, float* __restrict__ out,
                         int E, int Etot) {
  const int gid = blockIdx.x * blockDim.x + threadIdx.x;
  const int e = gid >> 5;
  if (e >= Etot) return;
  const int lane = gid & 31;
  int s, d; edge_sd(e, E, src, dst, &s, &d);
  const int head = lane >> 2;                         // (lane*4)/16
  const float alpha = exb[(size_t)e * 8 + head] * rden[d * 8 + head];
  const float4 v = *(const float4*)(h + (size_t)s * 128 + lane * 4);
  float* op = out + (size_t)d * 128 + lane * 4;
  atomicAdd(op + 0, v.x * alpha);
  atomicAdd(op + 1, v.y * alpha);
  atomicAdd(op + 2, v.z * alpha);
  atomicAdd(op + 3, v.w * alpha);
}

// F=40, H=1 (output layer): wave per edge, lanes stride the 40 channels.
__global__ void k_agg40(const float* __restrict__ h, const float* __restrict__ exb,
                        const float* __restrict__ rden, const int* __restrict__ src,
                        const int* __restrict__ dst, float* __restrict__ out,
                        int E, int Etot) {
  const int gid = blockIdx.x * blockDim.x + threadIdx.x;
  const int e = gid >> 5;
  if (e >= Etot) return;
  const int lane = gid & 31;
  int s, d; edge_sd(e, E, src, dst, &s, &d);
  const float alpha = exb[e] * rden[d];
  for (int c = lane; c < 40; c += 32) {
    atomicAdd(out + (size_t)d * 40 + c, h[(size_t)s * 40 + c] * alpha);
  }
}

__global__ void k_bias_act(float* __restrict__ p, const float* __restrict__ b,
                           int n, int F, int relu) {
  const int idx = blockIdx.x * blockDim.x + threadIdx.x;
  if (idx >= n) return;
  float v = p[idx] + b[idx % F];
  if (relu) v = fmaxf(v, 0.0f);
  p[idx] = v;
}

// ---------------------------------------------------------------------------
static void edge_phase(const float* hfeat, const float* a_s, const float* a_d,
                       float* als, float* ald, float* emax, float* den, float* exb,
                       const int* srcp, const int* dstp, float* out,
                       int N, int H, int C, int E, int Etot, hipStream_t stream) {
  const int F = H * C;
  const int tNH = N * H;
  k_logits<<<(tNH + 255) / 256, 256, 0, stream>>>(hfeat, a_s, a_d, als, ald, N, H, C);
  k_fill<<<(tNH + 255) / 256, 256, 0, stream>>>(emax, -INFINITY, tNH);
  hipMemsetAsync(den, 0, sizeof(float) * (size_t)tNH, stream);
  const int tE = Etot * H;
  k_edge_max<<<(tE + 255) / 256, 256, 0, stream>>>(als, ald, srcp, dstp, emax, E, Etot, H);
  k_edge_expsum<<<(tE + 255) / 256, 256, 0, stream>>>(als, ald, srcp, dstp, emax, den, exb, E, Etot, H);
  k_recip<<<(tNH + 255) / 256, 256, 0, stream>>>(den, tNH);
  hipMemsetAsync(out, 0, sizeof(float) * (size_t)N * F, stream);
  const int tA = Etot * 32;
  if (F == 128) {
    k_agg128<<<(tA + 255) / 256, 256, 0, stream>>>(hfeat, exb, den, srcp, dstp, out, E, Etot);
  } else {
    k_agg40<<<(tA + 255) / 256, 256, 0, stream>>>(hfeat, exb, den, srcp, dstp, out, E, Etot);
  }
}

static void gemm(const float* A, const float* B, float* C, int Nrows, int K, int Fout,
                 hipStream_t stream) {
  const int tiles_m = (Nrows + 15) / 16;
  const int tiles_n = (Fout + 15) / 16;
  const int threads = tiles_m * tiles_n * 32;
  k_gemm_wmma<<<(threads + 255) / 256, 256, 0, stream>>>(A, B, C, Nrows, K, Fout,
                                                         tiles_m, tiles_n);
}

extern "C" void kernel_launch(void* const* d_in, const int* in_sizes, int n_in,
                              void* d_out, int out_size, void* d_ws, size_t ws_size,
                              hipStream_t stream) {
  const float* x   = (const float*)d_in[0];
  const int*  eidx = (const int*)d_in[1];
  const float* W0  = (const float*)d_in[2];
  const float* as0 = (const float*)d_in[3];
  const float* ad0 = (const float*)d_in[4];
  const float* b0  = (const float*)d_in[5];
  const float* W1  = (const float*)d_in[6];
  const float* as1 = (const float*)d_in[7];
  const float* ad1 = (const float*)d_in[8];
  const float* b1  = (const float*)d_in[9];
  const float* W2  = (const float*)d_in[10];
  const float* as2 = (const float*)d_in[11];
  const float* ad2 = (const float*)d_in[12];
  const float* b2  = (const float*)d_in[13];

  const int F = 128, H = 8, C = 16, Cls = 40;
  const int N    = in_sizes[0] / F;
  const int E    = in_sizes[1] / 2;
  const int Etot = E + N;

  const int* srcp = eidx;
  const int* dstp = eidx + E;

  // workspace layout (floats): hA | hB | als | ald | emax | den | exb
  float* hA   = (float*)d_ws;                 // [N,128] pre-aggregation features
  float* hB   = hA   + (size_t)N * F;         // [N,128] aggregated / next input
  float* als  = hB   + (size_t)N * F;         // [N,H]
  float* ald  = als  + (size_t)N * H;         // [N,H]
  float* emax = ald  + (size_t)N * H;         // [N,H]
  float* den  = emax + (size_t)N * H;         // [N,H]
  float* exb  = den  + (size_t)N * H;         // [Etot,H]

  // ---- layer 0: 128 -> 8x16 ----
  gemm(x, W0, hA, N, F, F, stream);
  edge_phase(hA, as0, ad0, als, ald, emax, den, exb, srcp, dstp, hB,
             N, H, C, E, Etot, stream);
  k_bias_act<<<((N * F) + 255) / 256, 256, 0, stream>>>(hB, b0, N * F, F, 1);

  // ---- layer 1: 128 -> 8x16 ----
  gemm(hB, W1, hA, N, F, F, stream);
  edge_phase(hA, as1, ad1, als, ald, emax, den, exb, srcp, dstp, hB,
             N, H, C, E, Etot, stream);
  k_bias_act<<<((N * F) + 255) / 256, 256, 0, stream>>>(hB, b1, N * F, F, 1);

  // ---- layer 2: 128 -> 1x40 ----
  gemm(hB, W2, hA, N, F, Cls, stream);   // hA used as compact [N,40]
  edge_phase(hA, as2, ad2, als, ald, emax, den, exb, srcp, dstp, (float*)d_out,
             N, 1, Cls, E, Etot, stream);
  k_bias_act<<<((N * Cls) + 255) / 256, 256, 0, stream>>>((float*)d_out, b2, N * Cls, Cls, 0);
}